// LiquidLayer1_79809082294721
// MI455X (gfx1250) — compile-verified
//
#include <hip/hip_runtime.h>
#include <hip/hip_bf16.h>

// Problem constants (from reference): N=128 nodes, B=32 batch, T=256 steps,
// D=64 input dim, H=64 hidden. Output [B,T,N,H] f32.
#define NN 128
#define BB 32
#define TT 256
#define DD 64
#define HH 64

typedef __attribute__((ext_vector_type(2))) float v2f;
typedef __attribute__((ext_vector_type(8))) float v8f;

// ---------------------------------------------------------------------------
// Kernel 1: build the constant mixing matrix M = (I - tril(C,-1))^-1 (I+triu(C,1))
// by forward substitution. One block, 128 threads (thread = column), rows
// sequential with barriers. M kept in LDS (64KB) then flushed to workspace.
// ---------------------------------------------------------------------------
__global__ __launch_bounds__(128) void build_M_kernel(
    const unsigned char* __restrict__ conn, float* __restrict__ Mm)
{
    __shared__ float Ml[NN * NN];
    const int c = threadIdx.x;
    for (int i = 0; i < NN; ++i) {
        float a = (i == c) ? 1.f : 0.f;
        if (c > i && conn[i * NN + c]) a += 1.f;           // (I+U) row
        for (int j = 0; j < i; ++j) {                       // + L[i,j]*M[j,:]
            if (conn[i * NN + j]) a += Ml[j * NN + c];
        }
        Ml[i * NN + c] = a;
        __syncthreads();
    }
    for (int idx = c; idx < NN * NN; idx += 128) Mm[idx] = Ml[idx];
}

// ---------------------------------------------------------------------------
// Kernel 2: transpose W[n][h][k] -> Wt[n][k][h] (coalesced B-fragment loads
// in the cell GEMM), and zero the initial state S0.
// Blocks 0..127: transpose node n. Blocks 128..159: zero S0.
// ---------------------------------------------------------------------------
__global__ __launch_bounds__(256) void prep_kernel(
    const float* __restrict__ W, float* __restrict__ Wt, float* __restrict__ S0)
{
    const int bid = blockIdx.x;
    const int tid = threadIdx.x;
    if (bid < NN) {
        const float* Wn = W + bid * (HH * (DD + HH));   // [h][k]
        float*       Wo = Wt + bid * (HH * (DD + HH));  // [k][h]
        for (int o = tid; o < HH * (DD + HH); o += 256) {
            int k = o >> 6, h = o & 63;
            Wo[o] = Wn[h * (DD + HH) + k];
        }
    } else {
        float* p = S0 + (bid - NN) * 8192;
        for (int o = tid; o < 8192; o += 256) p[o] = 0.f;
    }
}

// ---------------------------------------------------------------------------
// Kernel 3: one timestep. Grid = 16 blocks (8 node-groups x 2 batch-groups),
// 256 threads = 8 wave32.
// Phase 1: eff[16n x (16b*64h)] = M[n0:,:] @ Sp  via v_wmma_f32_16x16x4_f32.
// Phase 2: per node, lin[16b x 64h] = [x_t | eff] @ Wt[n], fused gate/tanh
//          epilogue -> out and next state.
// ---------------------------------------------------------------------------
__global__ __launch_bounds__(256) void step_kernel(
    int t,
    const float* __restrict__ x,     // [B][T][D]
    const float* __restrict__ Mm,    // [N][N]
    const float* __restrict__ Wt,    // [N][D+H][H]
    const float* __restrict__ bvec,  // [N][H]
    const float* __restrict__ bias,  // [N][H]
    const float* __restrict__ decay, // [N][H]
    const float* __restrict__ iw,    // [N][H]
    const float* __restrict__ Sp,    // [N][B][H] state in
    float* __restrict__ Sn,          // [N][B][H] state out
    float* __restrict__ out)         // [B][T][N][H]
{
    __shared__ float eff[16 * 16 * 64]; // [node16][batch16][h64] = 64KB

    const int tid  = threadIdx.x;
    const int w    = tid >> 5;   // wave id 0..7
    const int lane = tid & 31;
    const int half = lane >> 4;  // 0/1
    const int l15  = lane & 15;
    const int n0 = (blockIdx.x & 7) * 16;  // node group base
    const int b0 = (blockIdx.x >> 3) * 16; // batch group base

    // ---------------- Phase 1: eff = M-rows @ Sp ----------------
    // A frag (16x4, f32): lane l15 -> row M (node), VGPR0 holds K = k0 + 2*half,
    // VGPR1 holds K = k0+1+2*half. B frag (4x16): lane l15 -> col N, same K map.
    v8f acc[8] = {};
    const float* Arow = Mm + (n0 + l15) * NN;
    for (int k0 = 0; k0 < NN; k0 += 4) {
        v2f a;
        a.x = Arow[k0 + 2 * half];
        a.y = Arow[k0 + 1 + 2 * half];
#pragma unroll
        for (int i = 0; i < 8; ++i) {
            const int nt = w * 8 + i;          // N-tile 0..63 over (b,h) cols
            const int bl = nt >> 2;            // local batch 0..15
            const int hb = (nt & 3) << 4;      // h base 0/16/32/48
            const float* Bp = Sp + (b0 + bl) * HH + hb + l15;
            v2f bf;
            bf.x = Bp[(k0 + 2 * half) * (BB * HH)];
            bf.y = Bp[(k0 + 1 + 2 * half) * (BB * HH)];
            acc[i] = __builtin_amdgcn_wmma_f32_16x16x4_f32(
                false, a, false, bf, (short)0, acc[i], false, false);
        }
    }
    // C/D layout: VGPR r -> node row r + 8*half; lane l15 -> col.
#pragma unroll
    for (int i = 0; i < 8; ++i) {
        const int nt = w * 8 + i;
        const int bl = nt >> 2;
        const int hb = (nt & 3) << 4;
#pragma unroll
        for (int r = 0; r < 8; ++r)
            eff[(r + 8 * half) * 1024 + bl * 64 + hb + l15] = acc[i][r];
    }
    __syncthreads();

    // ---------------- Phase 2: cell GEMM + gated epilogue ----------------
    // Each wave handles 2 nodes; per node: M=16 (batch) x N=64 (h) x K=128.
#pragma unroll
    for (int niLoop = 0; niLoop < 2; ++niLoop) {
        const int ni = w * 2 + niLoop;  // local node 0..15
        const int n  = n0 + ni;         // global node
        v8f c[4] = {};
        const float* xrow   = x + (b0 + l15) * (TT * DD) + t * DD; // K<64 source
        const float* effrow = eff + ni * 1024 + l15 * 64;          // K>=64 source
        const float* Wn     = Wt + n * (HH * (DD + HH));           // [k][h]
        for (int k0 = 0; k0 < DD + HH; k0 += 4) {
            v2f a;
            if (k0 < DD) {
                a.x = xrow[k0 + 2 * half];
                a.y = xrow[k0 + 1 + 2 * half];
            } else {
                a.x = effrow[k0 - DD + 2 * half];
                a.y = effrow[k0 - DD + 1 + 2 * half];
            }
#pragma unroll
            for (int j = 0; j < 4; ++j) {
                const int hb = j * 16;
                v2f bf;
                bf.x = Wn[(k0 + 2 * half) * HH + hb + l15];
                bf.y = Wn[(k0 + 1 + 2 * half) * HH + hb + l15];
                c[j] = __builtin_amdgcn_wmma_f32_16x16x4_f32(
                    false, a, false, bf, (short)0, c[j], false, false);
            }
        }
        // Epilogue: lin = acc + b + bias; gate = sigmoid(lin);
        // new = tanh(gate * (decay*eff + input_weight*lin))
#pragma unroll
        for (int j = 0; j < 4; ++j) {
            const int h  = j * 16 + l15;
            const float bb = bvec[n * HH + h] + bias[n * HH + h];
            const float dk = decay[n * HH + h];
            const float ww = iw[n * HH + h];
#pragma unroll
            for (int r = 0; r < 8; ++r) {
                const int m = r + 8 * half;   // local batch row
                const float lin  = c[j][r] + bb;
                const float gate = 1.f / (1.f + __expf(-lin));
                const float prev = eff[ni * 1024 + m * 64 + h];
                const float nv   = tanhf(gate * (dk * prev + ww * lin));
                out[((size_t)(b0 + m) * TT + t) * (NN * HH) + n * HH + h] = nv;
                Sn[n * (BB * HH) + (b0 + m) * HH + h] = nv;
            }
        }
    }
}

// ---------------------------------------------------------------------------
// Host: prep (M, Wt, S0) then 256 sequential step launches (graph-friendly).
// Workspace layout (floats): M[16384] | Wt[1048576] | S0[262144] | S1[262144]
// ---------------------------------------------------------------------------
extern "C" void kernel_launch(void* const* d_in, const int* in_sizes, int n_in,
                              void* d_out, int out_size, void* d_ws, size_t ws_size,
                              hipStream_t stream) {
    (void)in_sizes; (void)n_in; (void)out_size; (void)ws_size;
    const float* x     = (const float*)d_in[0];
    const float* W     = (const float*)d_in[1];
    const float* bvec  = (const float*)d_in[2];
    const float* bias  = (const float*)d_in[3];
    const float* decay = (const float*)d_in[4];
    const float* iw    = (const float*)d_in[5];
    const unsigned char* conn = (const unsigned char*)d_in[6]; // jnp.bool_ = 1 byte
    float* out = (float*)d_out;

    float* ws = (float*)d_ws;
    float* Mm = ws;
    float* Wt = Mm + NN * NN;
    float* S0 = Wt + NN * (DD + HH) * HH;
    float* S1 = S0 + NN * BB * HH;

    build_M_kernel<<<1, 128, 0, stream>>>(conn, Mm);
    prep_kernel<<<NN + 32, 256, 0, stream>>>(W, Wt, S0);
    for (int t = 0; t < TT; ++t) {
        const float* Sp = (t & 1) ? S1 : S0;
        float*       Sn = (t & 1) ? S0 : S1;
        step_kernel<<<16, 256, 0, stream>>>(t, x, Mm, Wt, bvec, bias, decay, iw,
                                            Sp, Sn, out);
    }
}